// OuterLaplaceAggregation_5772436045848
// MI455X (gfx1250) — compile-verified
//
#include <hip/hip_runtime.h>
#include <math.h>
#include <stdint.h>

// Problem constants (from reference)
#define B_DIM   8
#define T_LEN   512
#define F_DIM   1024
#define C_DIM   4
// LIMIT = log(FLT_MAX)/1024 - 0.01
#define LIMIT_F 0.07664339751178551f

// Tiling
#define TPB     256          // threads per block = f-tile width
#define CHUNK   64           // output timesteps per block
#define NCHUNK  (T_LEN / CHUNK)   // 8
#define TS      32           // timesteps staged into LDS per async tile

typedef __attribute__((ext_vector_type(4))) int   v4i;
typedef __attribute__((ext_vector_type(4))) float v4f;

typedef __attribute__((address_space(1))) v4i* gptr_v4i;
typedef __attribute__((address_space(3))) v4i* lptr_v4i;

// ---------------- CDNA5 async global->LDS helpers -------------------------

__device__ __forceinline__ void async_copy_b128(const float* gsrc, float* ldst) {
#if __has_builtin(__builtin_amdgcn_global_load_async_to_lds_b128)
  __builtin_amdgcn_global_load_async_to_lds_b128(
      (gptr_v4i)(gsrc), (lptr_v4i)(ldst), /*offset=*/0, /*cpol=*/0);
#else
  unsigned loff = (unsigned)(uintptr_t)(__attribute__((address_space(3))) void*)(ldst);
  asm volatile("global_load_async_to_lds_b128 %0, %1, off"
               :: "v"(loff), "v"(gsrc) : "memory");
#endif
}

// wait until at most N async ops outstanding (in-order completion)
__device__ __forceinline__ void async_wait_8() {
#if __has_builtin(__builtin_amdgcn_s_wait_asynccnt)
  __builtin_amdgcn_s_wait_asynccnt(8);
#else
  asm volatile("s_wait_asynccnt 8" ::: "memory");
#endif
}
__device__ __forceinline__ void async_wait_0() {
#if __has_builtin(__builtin_amdgcn_s_wait_asynccnt)
  __builtin_amdgcn_s_wait_asynccnt(0);
#else
  asm volatile("s_wait_asynccnt 0" ::: "memory");
#endif
}

// ---------------- scan kernel ---------------------------------------------
// out[b,t,f,c] = h[t] where h[-1] = mem[b,f,c], h[t] = lambda*h[t-1] + x[b,t,f]
// lambda = exp(clip(a_f)) * (cos b_c + i sin b_c)

__global__ __launch_bounds__(TPB)
void laplace_scan_kernel(const float* __restrict__ x,
                         const float* __restrict__ mem_r,
                         const float* __restrict__ mem_i,
                         const float* __restrict__ a,
                         const float* __restrict__ bvec,
                         float* __restrict__ out) {
  __shared__ float xs[2][TS * TPB];   // 2 x 32 KB double-buffered staging tiles

  const int f0  = blockIdx.x * TPB;
  const int b   = blockIdx.y;
  const int g   = blockIdx.z;           // time chunk index
  const int tid = threadIdx.x;
  const int f   = f0 + tid;

  // per-thread decay factors
  float af = a[f];
  af = fminf(fmaxf(af, -LIMIT_F), -1e-8f);
  const float e = expf(af);
  float lr[C_DIM], li[C_DIM];
#pragma unroll
  for (int c = 0; c < C_DIM; ++c) {
    const float bc = bvec[c];
    lr[c] = e * cosf(bc);
    li[c] = e * sinf(bc);
  }

  // initial state = memory (shape [B,1,F,C])
  float hr[C_DIM], hi[C_DIM];
#pragma unroll
  for (int c = 0; c < C_DIM; ++c) {
    hr[c] = mem_r[((size_t)b * F_DIM + f) * C_DIM + c];
    hi[c] = mem_i[((size_t)b * F_DIM + f) * C_DIM + c];
  }

  const int t_main  = g * CHUNK;               // first timestep this block stores
  const int ntiles  = (t_main + CHUNK) / TS;   // tiles to scan (prefix + chunk)
  const int mtile0  = t_main / TS;             // first tile that stores output
  const float* xb   = x + (size_t)b * T_LEN * F_DIM + f0;  // [t][f-tile]

  // async staging geometry: each thread issues TS/4 = 8 b128 copies
  // (4 consecutive f for one t-row); the 8 waves cover a TS x TPB tile.
  const int fq  = (tid & 63) * 4;   // 64 quads cover 256 f
  const int tr0 = tid >> 6;         // 0..3

  auto stage = [&](int tile, int buf) {
    const float* base = xb + (size_t)tile * TS * F_DIM;
#pragma unroll
    for (int j = 0; j < TS / 4; ++j) {
      const int trow = tr0 + j * 4;
      async_copy_b128(base + (size_t)trow * F_DIM + fq,
                      &xs[buf][trow * TPB + fq]);
    }
  };

  stage(0, 0);                         // prime the pipeline (8 async ops)

  for (int k = 0; k < ntiles; ++k) {
    if (k + 1 < ntiles) {
      // refill the other buffer (consumed >= 1 full barrier ago), then wait
      // for the 8 oldest ops (tile k) -- async loads complete in order.
      stage(k + 1, (k + 1) & 1);
      async_wait_8();
    } else {
      async_wait_0();
    }
    __syncthreads();                   // tile k visible to all waves

    const float* buf = xs[k & 1];
    if (k >= mtile0) {
      // ---- stored tile ----
      const int tbase = k * TS;
#pragma unroll 8
      for (int j = 0; j < TS; ++j) {
        const float xv = buf[j * TPB + tid];
#pragma unroll
        for (int c = 0; c < C_DIM; ++c) {
          const float nr = fmaf(lr[c], hr[c], fmaf(-li[c], hi[c], xv));
          const float ni = fmaf(lr[c], hi[c], li[c] * hr[c]);
          hr[c] = nr; hi[c] = ni;
        }
        // out[b,t,f,c] complex64 interleaved: 8 floats contiguous per (t,f)
        v4f* op = (v4f*)(out + (((size_t)b * T_LEN + (tbase + j)) * F_DIM + f) * (C_DIM * 2));
        v4f o0 = {hr[0], hi[0], hr[1], hi[1]};
        v4f o1 = {hr[2], hi[2], hr[3], hi[3]};
        __builtin_nontemporal_store(o0, op + 0);
        __builtin_nontemporal_store(o1, op + 1);
      }
    } else {
      // ---- prefix tile: advance state only ----
#pragma unroll 8
      for (int j = 0; j < TS; ++j) {
        const float xv = buf[j * TPB + tid];
#pragma unroll
        for (int c = 0; c < C_DIM; ++c) {
          const float nr = fmaf(lr[c], hr[c], fmaf(-li[c], hi[c], xv));
          const float ni = fmaf(lr[c], hi[c], li[c] * hr[c]);
          hr[c] = nr; hi[c] = ni;
        }
      }
    }
    __syncthreads();                   // all waves done with buf before refill
  }
}

// ---------------- launch ---------------------------------------------------

extern "C" void kernel_launch(void* const* d_in, const int* in_sizes, int n_in,
                              void* d_out, int out_size, void* d_ws, size_t ws_size,
                              hipStream_t stream) {
  const float* x     = (const float*)d_in[0];   // [8,512,1024,1]
  const float* mem_r = (const float*)d_in[1];   // [8,1,1024,4]
  const float* mem_i = (const float*)d_in[2];   // [8,1,1024,4]
  const float* a     = (const float*)d_in[3];   // [1024]
  const float* bvec  = (const float*)d_in[4];   // [4]
  float* out         = (float*)d_out;           // [8,512,1024,4] complex64 -> interleaved f32

  dim3 grid(F_DIM / TPB, B_DIM, NCHUNK);        // 4 x 8 x 8 = 256 blocks
  dim3 block(TPB);                              // 256 threads = 8 wave32
  laplace_scan_kernel<<<grid, block, 0, stream>>>(x, mem_r, mem_i, a, bvec, out);
}